// predict_head_25984552141160
// MI455X (gfx1250) — compile-verified
//
#include <hip/hip_runtime.h>
#include <math.h>

// ---------------- problem constants ----------------
constexpr int BB = 64;          // batch
constexpr int AA = 16;          // attention heads / "A"
constexpr int CC = 512;         // channels
constexpr int HH = 30, WW = 40;
constexpr int HWX = HH * WW;    // 1200
constexpr int EE = 256;
constexpr int PP = 512;

// hw tiling for the fused kernel: 1200 = 5 chunks * 5 tiles * 48
constexpr int NSPLIT = 5;       // chunks per batch
constexpr int TILES  = 5;       // tiles per chunk
constexpr int TILE   = 48;      // hw positions per tile
constexpr int CHUNK  = TILES * TILE;   // 240

// d_out layout (floats), reference tuple order:
// aggr_z (64*1201), mu (64), sigma2 (64), aggr_x (64*1200), x (64*16*1200)
constexpr size_t OFF_AGGRZ = 0;
constexpr size_t OFF_MU    = OFF_AGGRZ + (size_t)BB * (1 + HWX);   // 76864
constexpr size_t OFF_S2    = OFF_MU + BB;                          // 76928
constexpr size_t OFF_AGGRX = OFF_S2 + BB;                          // 76992
constexpr size_t OFF_X     = OFF_AGGRX + (size_t)BB * HWX;         // 153792

// workspace layout (floats)
constexpr size_t WS_FEATSUM = 0;                                    // 64*512
constexpr size_t WS_AGGRP   = WS_FEATSUM + (size_t)BB * CC;         // 64*5*16*512
constexpr size_t WS_ATTW    = WS_AGGRP + (size_t)BB * NSPLIT * AA * CC;

typedef float v2f __attribute__((ext_vector_type(2)));
typedef float v8f __attribute__((ext_vector_type(8)));

__device__ __forceinline__ v8f wmma_f32(v2f a, v2f b, v8f c) {
  // D = A(16x4 f32) * B(4x16 f32) + C(16x16 f32)
  return __builtin_amdgcn_wmma_f32_16x16x4_f32(false, a, false, b, (short)0, c,
                                               false, false);
}

// padded LDS strides (floats) chosen for conflict-free strided fragment reads
constexpr int W3S_STRIDE = 516;   // 516 mod 64 banks -> distinct, 16B-aligned rows
constexpr int XS_STRIDE  = 52;    // 52*lane mod 64 distinct for lane<16

// =====================================================================
// Kernel D: duration branch (7x7/s5 conv -> 6x8 -> (6,8) conv -> mu, exp)
//           + featsum[b,c] = sum_hw features[b,c,:]
// Run FIRST: streams `features` once, warming L2 (157MB < 192MB L2) for kA.
// =====================================================================
__global__ __launch_bounds__(128) void kD_drt_featsum(
    const float* __restrict__ feat, const float* __restrict__ w1,
    const float* __restrict__ bb1, const float* __restrict__ w2,
    const float* __restrict__ bb2, float* __restrict__ featsum,
    float* __restrict__ mu_out, float* __restrict__ s2_out) {
  const int b = blockIdx.x;
  const int tid = threadIdx.x;
  const int lane = tid & 31, wv = tid >> 5;
  const float* fb = feat + (size_t)b * CC * HWX;

  __shared__ float red[96];
  __shared__ float t1s[48];

  // ---- featsum: wave wv owns channels [wv*128, wv*128+128) ----
  for (int c = wv * 128; c < wv * 128 + 128; ++c) {
    const float* row = fb + (size_t)c * HWX;
    float acc = 0.f;
    for (int p = lane; p < HWX; p += 32) acc += row[p];
    acc += __shfl_xor(acc, 1);
    acc += __shfl_xor(acc, 2);
    acc += __shfl_xor(acc, 4);
    acc += __shfl_xor(acc, 8);
    acc += __shfl_xor(acc, 16);
    if (lane == 0) featsum[b * CC + c] = acc;
  }

  // ---- conv1 partials: 48 outputs x 2 channel-halves = 96 threads ----
  if (tid < 96) {
    const int out = tid % 48, half = tid / 48;
    const int oh = out >> 3, ow = out & 7;
    const int ih0 = oh * 5 - 2, iw0 = ow * 5 - 2;
    float acc = 0.f;
    for (int c = half * 256; c < half * 256 + 256; ++c) {
      const float* row = fb + (size_t)c * HWX;
      const float* wr = w1 + c * 49;
      #pragma unroll
      for (int kh = 0; kh < 7; ++kh) {
        const int ih = ih0 + kh;
        if ((unsigned)ih >= (unsigned)HH) continue;
        #pragma unroll
        for (int kw = 0; kw < 7; ++kw) {
          const int iw = iw0 + kw;
          if ((unsigned)iw >= (unsigned)WW) continue;
          acc += row[ih * WW + iw] * wr[kh * 7 + kw];
        }
      }
    }
    red[half * 48 + out] = acc;
  }
  __syncthreads();
  if (tid < 48) t1s[tid] = fmaxf(red[tid] + red[48 + tid] + bb1[0], 0.f);
  __syncthreads();
  if (tid < 2) {
    float acc = bb2[tid];
    #pragma unroll 8
    for (int i = 0; i < 48; ++i) acc += t1s[i] * w2[tid * 48 + i];
    if (tid == 0) mu_out[b] = acc;
    else          s2_out[b] = expf(acc);
  }
}

// =====================================================================
// Kernel A: fused  x = relu(W3 @ F + b3)  and  aggr += x @ E^T
// grid (64 batches, 5 chunks), 128 threads (4 waves).
// Per 48-wide tile: waves 0..2 each produce one 16x16 x-tile via
// V_WMMA_F32_16X16X4_F32 (K=512); wave fragments: A from LDS w3
// (padded stride), B = coalesced global feature reads. Then all 4 waves
// (128 channels each) accumulate aggr with x as A (from LDS) and
// encoder_feature as B (aligned b64 global reads, line-reuse across K).
// =====================================================================
__global__ __launch_bounds__(128) void kA_x_aggr(
    const float* __restrict__ enc, const float* __restrict__ feat,
    const float* __restrict__ w3, const float* __restrict__ b3,
    float* __restrict__ xout, float* __restrict__ aggrp) {
  const int b = blockIdx.x, s = blockIdx.y;
  const int tid = threadIdx.x;
  const int lane = tid & 31, wv = tid >> 5;
  const int lo = lane & 15, hi = lane >> 4;

  __shared__ float w3s[AA * W3S_STRIDE];   // 33 KB
  __shared__ float xs[AA * XS_STRIDE];     // 3.3 KB

  // stage w_sal3 (16x512) with padded stride
  for (int i = tid; i < AA * CC; i += 128)
    w3s[(i >> 9) * W3S_STRIDE + (i & 511)] = w3[i];

  float vb[8];
  #pragma unroll
  for (int r = 0; r < 8; ++r) vb[r] = b3[r + 8 * hi];
  __syncthreads();

  const size_t fbase = (size_t)b * CC * HWX;
  const int aoff  = lo * W3S_STRIDE + 2 * hi;   // A-frag base into w3s
  const int axoff = lo * XS_STRIDE + 2 * hi;    // A-frag base into xs
  const int c0 = wv * 128;                      // this wave's channel range

  v8f acc8[8];
  #pragma unroll
  for (int n = 0; n < 8; ++n) {
    #pragma unroll
    for (int r = 0; r < 8; ++r) acc8[n][r] = 0.f;
  }

  for (int t = 0; t < TILES; ++t) {
    const int hw0 = s * CHUNK + t * TILE;

    // ---- phase 1: x-tile GEMM (waves 0..2), wave 3 idles ----
    if (wv < 3) {
      const int n0 = wv * 16;
      v8f acc;
      #pragma unroll
      for (int r = 0; r < 8; ++r) acc[r] = 0.f;
      // B stream: rows k+2*hi / k+1+2*hi, cols hw0+n0+lo (half-wave coalesced)
      const float* fB = feat + fbase + (size_t)(2 * hi) * HWX + hw0 + n0 + lo;
      #pragma unroll 4
      for (int k = 0; k < CC; k += 4) {
        v2f af = *(const v2f*)(w3s + aoff + k);     // ds_load_b64
        v2f bf;
        bf.x = fB[0];
        bf.y = fB[HWX];
        fB += 4 * HWX;
        acc = wmma_f32(af, bf, acc);
      }
      #pragma unroll
      for (int r = 0; r < 8; ++r) {
        const float v = fmaxf(acc[r] + vb[r], 0.f);
        const int ai = r + 8 * hi;
        const int col = n0 + lo;
        xs[ai * XS_STRIDE + col] = v;
        xout[((size_t)b * AA + ai) * HWX + hw0 + col] = v;
      }
    }
    __syncthreads();

    // ---- phase 2: aggr += x @ E^T  (all 4 waves, disjoint channels) ----
    v2f af[12];
    #pragma unroll
    for (int kk = 0; kk < 12; ++kk)
      af[kk] = *(const v2f*)(xs + axoff + kk * 4);  // ds_load_b64

    #pragma unroll
    for (int n = 0; n < 8; ++n) {
      const int cc = c0 + n * 16 + lo;
      const float* eB = enc + fbase + (size_t)cc * HWX + hw0 + 2 * hi;
      #pragma unroll 4
      for (int kk = 0; kk < 12; ++kk) {
        v2f bf = *(const v2f*)(eB + kk * 4);        // global_load_b64
        acc8[n] = wmma_f32(af[kk], bf, acc8[n]);
      }
    }
    __syncthreads();   // xs reused next tile
  }

  // write aggr partials (scaled by 1/HW later, reduced in kB)
  #pragma unroll
  for (int n = 0; n < 8; ++n) {
    const int cc = c0 + n * 16 + lo;
    #pragma unroll
    for (int r = 0; r < 8; ++r) {
      const int ai = r + 8 * hi;
      aggrp[(((size_t)b * NSPLIT + s) * AA + ai) * CC + cc] = acc8[n][r];
    }
  }
}

// =====================================================================
// Kernel B: reduce aggr partials, proj_f GEMM (WMMA), tanh/att scores,
// softmax over A=16, attw -> ws, aggr_z[b,0] -> d_out.  One WG per batch.
// =====================================================================
__global__ __launch_bounds__(128) void kB_attention(
    const float* __restrict__ aggrp, const float* __restrict__ featsum,
    const float* __restrict__ subj, const float* __restrict__ w_sal2,
    const float* __restrict__ b_sal2, const float* __restrict__ w_pf,
    const float* __restrict__ b_pf, const float* __restrict__ w_ps,
    const float* __restrict__ b_ps, const float* __restrict__ w_att,
    const float* __restrict__ b_att, float* __restrict__ attw,
    float* __restrict__ aggrz) {
  const int b = blockIdx.x;
  const int tid = threadIdx.x;
  const int lane = tid & 31, wv = tid >> 5;
  const int lo = lane & 15, hi = lane >> 4;

  __shared__ float ag[AA * W3S_STRIDE];  // reduced aggr_feature, padded
  __shared__ float ps[PP];               // proj_s + b_ps + b_pf
  __shared__ float ypart[128];
  __shared__ float ysh[AA];
  __shared__ float sc[AA];

  // reduce the 5 split partials, apply 1/HW
  for (int i = tid; i < AA * CC; i += 128) {
    const int a = i >> 9, c = i & 511;
    float v = 0.f;
    #pragma unroll
    for (int sp = 0; sp < NSPLIT; ++sp)
      v += aggrp[(((size_t)b * NSPLIT + sp) * AA + a) * CC + c];
    ag[a * W3S_STRIDE + c] = v * (1.f / (float)HWX);
  }

  // proj_s (+ both biases folded in)
  for (int p = tid; p < PP; p += 128) {
    float v = b_ps[p] + b_pf[p];
    const float* wp = w_ps + (size_t)p * EE;
    const float* sb = subj + b * EE;
    #pragma unroll 8
    for (int e = 0; e < EE; ++e) v += sb[e] * wp[e];
    ps[p] = v;
  }

  // y[a] = featsum/HW @ w_sal2 + b_sal2
  {
    const int a = tid >> 3, part = tid & 7;
    float v = 0.f;
    const float* fs = featsum + b * CC;
    const float* w2 = w_sal2 + (size_t)a * CC;
    for (int c = part * 64; c < part * 64 + 64; ++c) v += fs[c] * w2[c];
    ypart[tid] = v;
  }
  if (tid < AA) sc[tid] = b_att[0];
  __syncthreads();
  if (tid < AA) {
    float v = 0.f;
    #pragma unroll
    for (int j = 0; j < 8; ++j) v += ypart[tid * 8 + j];
    ysh[tid] = v * (1.f / (float)HWX) + b_sal2[tid];
  }
  __syncthreads();

  // proj_f GEMM (16 x 512 x 512) fused with tanh + w_att reduction
  const int aoff = lo * W3S_STRIDE + 2 * hi;
  float scacc[8];
  #pragma unroll
  for (int r = 0; r < 8; ++r) scacc[r] = 0.f;

  for (int n = 0; n < 8; ++n) {
    const int p = wv * 128 + n * 16 + lo;
    v8f acc;
    #pragma unroll
    for (int r = 0; r < 8; ++r) acc[r] = 0.f;
    const float* wb = w_pf + (size_t)p * CC + 2 * hi;
    #pragma unroll 4
    for (int k = 0; k < CC; k += 4) {
      v2f af = *(const v2f*)(ag + aoff + k);   // ds_load_b64
      v2f bf = *(const v2f*)(wb + k);          // global_load_b64 (contig K)
      acc = wmma_f32(af, bf, acc);
    }
    const float pv = ps[p];
    const float wa = w_att[p];
    #pragma unroll
    for (int r = 0; r < 8; ++r) scacc[r] += wa * tanhf(acc[r] + pv);
  }
  // reduce the 16 lanes sharing the same 'a' row, then LDS-atomic once
  #pragma unroll
  for (int r = 0; r < 8; ++r) {
    float v = scacc[r];
    v += __shfl_xor(v, 1);
    v += __shfl_xor(v, 2);
    v += __shfl_xor(v, 4);
    v += __shfl_xor(v, 8);
    if (lo == 0) atomicAdd(&sc[r + 8 * hi], v);
  }
  __syncthreads();

  if (tid == 0) {
    float m = -3.4e38f;
    #pragma unroll
    for (int a = 0; a < AA; ++a) m = fmaxf(m, sc[a]);
    float e[AA], sum = 0.f;
    #pragma unroll
    for (int a = 0; a < AA; ++a) { e[a] = expf(sc[a] - m); sum += e[a]; }
    const float inv = 1.f / sum;
    float az0 = 0.f;
    #pragma unroll
    for (int a = 0; a < AA; ++a) {
      const float w = e[a] * inv;
      attw[b * AA + a] = w;
      az0 += ysh[a] * w;
    }
    aggrz[(size_t)b * (1 + HWX)] = az0;   // aggr_z[b, 0] (y part)
  }
}

// =====================================================================
// Kernel C: aggr_x[b,hw] = sum_a attw[b,a] * x[b,a,hw]
//           (also equals aggr_z[b, 1+hw])
// =====================================================================
__global__ __launch_bounds__(128) void kC_aggr_out(
    const float* __restrict__ x, const float* __restrict__ attw,
    float* __restrict__ aggrx, float* __restrict__ aggrz) {
  const int b = blockIdx.x;
  const int hw = blockIdx.y * 128 + threadIdx.x;
  __shared__ float aw[AA];
  if (threadIdx.x < AA) aw[threadIdx.x] = attw[b * AA + threadIdx.x];
  __syncthreads();
  if (hw < HWX) {
    const float* xb = x + (size_t)b * AA * HWX + hw;
    float acc = 0.f;
    #pragma unroll
    for (int a = 0; a < AA; ++a) acc += xb[(size_t)a * HWX] * aw[a];
    aggrx[(size_t)b * HWX + hw] = acc;
    aggrz[(size_t)b * (1 + HWX) + 1 + hw] = acc;
  }
}

// =====================================================================
extern "C" void kernel_launch(void* const* d_in, const int* in_sizes, int n_in,
                              void* d_out, int out_size, void* d_ws,
                              size_t ws_size, hipStream_t stream) {
  const float* enc    = (const float*)d_in[0];
  const float* feat   = (const float*)d_in[1];
  const float* subj   = (const float*)d_in[2];
  const float* w_sal2 = (const float*)d_in[3];
  const float* b_sal2 = (const float*)d_in[4];
  const float* w_sal3 = (const float*)d_in[5];
  const float* b_sal3 = (const float*)d_in[6];
  const float* w_drt1 = (const float*)d_in[7];
  const float* b_drt1 = (const float*)d_in[8];
  const float* w_drt2 = (const float*)d_in[9];
  const float* b_drt2 = (const float*)d_in[10];
  const float* w_pf   = (const float*)d_in[11];
  const float* b_pf   = (const float*)d_in[12];
  const float* w_ps   = (const float*)d_in[13];
  const float* b_ps   = (const float*)d_in[14];
  const float* w_att  = (const float*)d_in[15];
  const float* b_att  = (const float*)d_in[16];

  float* out    = (float*)d_out;
  float* aggrz  = out + OFF_AGGRZ;
  float* mu     = out + OFF_MU;
  float* s2     = out + OFF_S2;
  float* aggrx  = out + OFF_AGGRX;
  float* x      = out + OFF_X;

  float* ws      = (float*)d_ws;
  float* featsum = ws + WS_FEATSUM;   // 64*512
  float* aggrp   = ws + WS_AGGRP;     // 64*5*16*512
  float* attw    = ws + WS_ATTW;      // 64*16

  // 1) duration branch + featsum (streams `features`, warms 192MB L2)
  kD_drt_featsum<<<BB, 128, 0, stream>>>(feat, w_drt1, b_drt1, w_drt2, b_drt2,
                                         featsum, mu, s2);
  // 2) fused x-GEMM + aggr-GEMM (WMMA f32), the only pass over encoder_feature
  kA_x_aggr<<<dim3(BB, NSPLIT), 128, 0, stream>>>(enc, feat, w_sal3, b_sal3, x,
                                                  aggrp);
  // 3) reduce + proj_f WMMA + attention softmax
  kB_attention<<<BB, 128, 0, stream>>>(aggrp, featsum, subj, w_sal2, b_sal2,
                                       w_pf, b_pf, w_ps, b_ps, w_att, b_att,
                                       attw, aggrz);
  // 4) attention-weighted outputs
  kC_aggr_out<<<dim3(BB, 10), 128, 0, stream>>>(x, attw, aggrx, aggrz);
}